// SpatialTransformer_17128329576936
// MI455X (gfx1250) — compile-verified
//
#include <hip/hip_runtime.h>

typedef __attribute__((ext_vector_type(2))) float v2f;
typedef __attribute__((ext_vector_type(8))) float v8f;

#define DD 160
#define HH 192
#define WW 160
#define HW (HH * WW)      // 30720
#define VOL (DD * HW)     // 4915200 ; divisible by 256 -> no tail, EXEC stays all-1s
#define CD 79.5f
#define CH 95.5f
#define CW 79.5f

__global__ __launch_bounds__(256) void st_affine_wmma_kernel(
    const float* __restrict__ vol,   // [B, D, H, W, 1] f32
    const float* __restrict__ trf,   // [B, 4, 4] f32
    float* __restrict__ out)         // [B, D, H, W, 1] f32
{
    const int b    = blockIdx.y;
    const int lane = threadIdx.x & 31;
    const int m    = lane & 15;
    const bool hi  = lane >= 16;
    // wave-base voxel index within this batch's volume (32 consecutive voxels/wave)
    const int base = blockIdx.x * 256 + (threadIdx.x & ~31);

    const float* A = trf + b * 16;

    // ---- A-matrix fragment: 16x4 f32, rows 0..2 = affine rows, center folded into col 3.
    // Layout (ISA 7.12.2, 32-bit A 16x4): lane<16 -> {K0,K1}; lane>=16 -> {K2,K3}.
    float a0 = 0.0f, a1 = 0.0f;
    if (m < 3) {
        const float c0  = A[m * 4 + 0];
        const float c1  = A[m * 4 + 1];
        const float c2  = A[m * 4 + 2];
        const float ctr = (m == 0) ? CD : (m == 1) ? CH : CW;
        const float c3  = A[m * 4 + 3] + ctr;     // + center[m] via homogeneous 1 in B row 3
        a0 = hi ? c2 : c0;
        a1 = hi ? c3 : c1;
    }
    v2f amat; amat.x = a0; amat.y = a1;

    // ---- B fragments: two 4x16 groups of voxel columns [xc, yc, zc, 1].
    // Layout mirrors A: lane<16 -> rows {0,1} (col = lane); lane>=16 -> rows {2,3} (col = lane-16).
    v2f bmat[2];
#pragma unroll
    for (int g = 0; g < 2; ++g) {
        const int vidx = base + g * 16 + m;       // voxel for column m of group g
        const int dz   = vidx / HW;
        const int rem  = vidx - dz * HW;
        const int hy   = rem / WW;
        const int wx   = rem - hy * WW;
        const float xc = (float)dz - CD;
        const float yc = (float)hy - CH;
        const float zc = (float)wx - CW;
        bmat[g].x = hi ? zc   : xc;
        bmat[g].y = hi ? 1.0f : yc;
    }

    // ---- D[m][n] = sample coordinate m of voxel (base + g*16 + n); lanes 0-15 hold rows 0-7.
    v8f acc0 = {};
    v8f acc1 = {};
    acc0 = __builtin_amdgcn_wmma_f32_16x16x4_f32(false, amat, false, bmat[0],
                                                 (short)0, acc0, false, false);
    acc1 = __builtin_amdgcn_wmma_f32_16x16x4_f32(false, amat, false, bmat[1],
                                                 (short)0, acc1, false, false);

    // lane j (<16) owns voxel base+j   -> coords in acc0[0..2]
    // lane 16+j    owns voxel base+16+j -> coords sit in lane j of acc1; route via bpermute
    const float g1x = __shfl(acc1[0], m, 32);
    const float g1y = __shfl(acc1[1], m, 32);
    const float g1z = __shfl(acc1[2], m, 32);
    const float sx = hi ? g1x : acc0[0];
    const float sy = hi ? g1y : acc0[1];
    const float sz = hi ? g1z : acc0[2];

    // ---- trilinear interpolation, faithful to the reference clamp scheme
    const float maxx = (float)(DD - 1), maxy = (float)(HH - 1), maxz = (float)(WW - 1);
    const float fx = floorf(sx), fy = floorf(sy), fz = floorf(sz);
    const float cx = fminf(fmaxf(sx, 0.0f), maxx);
    const float cy = fminf(fmaxf(sy, 0.0f), maxy);
    const float cz = fminf(fmaxf(sz, 0.0f), maxz);
    const float x0 = fminf(fmaxf(fx, 0.0f), maxx);
    const float y0 = fminf(fmaxf(fy, 0.0f), maxy);
    const float z0 = fminf(fmaxf(fz, 0.0f), maxz);
    const float x1 = fminf(x0 + 1.0f, maxx);
    const float y1 = fminf(y0 + 1.0f, maxy);
    const float z1 = fminf(z0 + 1.0f, maxz);

    const float wx0 = x1 - cx, wy0 = y1 - cy, wz0 = z1 - cz;   // d1: lower-corner weights
    const float wx1 = 1.0f - wx0, wy1 = 1.0f - wy0, wz1 = 1.0f - wz0;

    const int ix0 = (int)x0, iy0 = (int)y0, iz0 = (int)z0;
    const int ix1 = (int)x1, iy1 = (int)y1, iz1 = (int)z1;

    const float* vb = vol + (size_t)b * VOL;
    const int px0 = ix0 * HW, px1 = ix1 * HW;
    const int py0 = iy0 * WW, py1 = iy1 * WW;

    const float v000 = vb[px0 + py0 + iz0];
    const float v001 = vb[px0 + py0 + iz1];
    const float v010 = vb[px0 + py1 + iz0];
    const float v011 = vb[px0 + py1 + iz1];
    const float v100 = vb[px1 + py0 + iz0];
    const float v101 = vb[px1 + py0 + iz1];
    const float v110 = vb[px1 + py1 + iz0];
    const float v111 = vb[px1 + py1 + iz1];

    const float r =
        wx0 * (wy0 * (wz0 * v000 + wz1 * v001) +
               wy1 * (wz0 * v010 + wz1 * v011)) +
        wx1 * (wy0 * (wz0 * v100 + wz1 * v101) +
               wy1 * (wz0 * v110 + wz1 * v111));

    out[(size_t)b * VOL + base + lane] = r;
}

extern "C" void kernel_launch(void* const* d_in, const int* in_sizes, int n_in,
                              void* d_out, int out_size, void* d_ws, size_t ws_size,
                              hipStream_t stream) {
    (void)in_sizes; (void)n_in; (void)out_size; (void)d_ws; (void)ws_size;
    const float* vol = (const float*)d_in[0];   // [4,160,192,160,1] f32
    const float* trf = (const float*)d_in[1];   // [4,4,4] f32
    float*       out = (float*)d_out;           // [4,160,192,160,1] f32

    dim3 grid(VOL / 256, 4, 1);                 // 19200 x 4 blocks, exact fit
    st_affine_wmma_kernel<<<grid, dim3(256, 1, 1), 0, stream>>>(vol, trf, out);
}